// ACmix_54846732370256
// MI455X (gfx1250) — compile-verified
//
#include <hip/hip_runtime.h>
#include <math.h>

typedef __attribute__((ext_vector_type(2))) float v2f;
typedef __attribute__((ext_vector_type(8))) float v8f;

#define NB    2
#define CCH   64          // Cin == Cout == 64
#define CD    16          // depth "c"
#define HH    20
#define WW    20
#define HWSP  (HH*WW)     // 400
#define CHW   (CD*HWSP)   // 6400
#define HD    16          // head dim
#define KATAP 7
#define KKA   49
#define PADR  3

// ---------------------------------------------------------------------------
// QKV 1x1x1 projections as WMMA f32 16x16x4 GEMMs.
// Y[b][o][s] = sum_i W[o][i] * X[b][i][s] + bias[o],  o,i in [0,64), s in [0,6400)
// One wave computes one 16(M=o) x 16(N=s) tile; K=64 reduced in 16 steps of 4.
// A frag (16x4 f32): lanes 0-15 hold M=lane, K={0,1}; lanes 16-31 M=lane-16, K={2,3}.
// B frag (4x16 f32): lanes 0-15 hold N=lane, K={0,1}; lanes 16-31 N=lane-16, K={2,3}.
// D (16x16 f32): VGPR r, lane<16 -> M=r,N=lane; lane>=16 -> M=r+8,N=lane-16.
// ---------------------------------------------------------------------------
__global__ void qkv_proj_wmma(const float* __restrict__ x,
                              const float* __restrict__ wq, const float* __restrict__ bq,
                              const float* __restrict__ wk, const float* __restrict__ bk,
                              const float* __restrict__ wv, const float* __restrict__ bv,
                              float* __restrict__ q, float* __restrict__ k,
                              float* __restrict__ v)
{
  const int NT     = CHW / 16;        // 400 N-tiles
  const int perMat = NB * 4 * NT;     // 3200 tiles per matrix
  int wave = (blockIdx.x * blockDim.x + threadIdx.x) >> 5;
  int lane = threadIdx.x & 31;
  int mat  = wave / perMat;
  int rem  = wave - mat * perMat;
  int bb   = rem / (4 * NT);
  int rem2 = rem - bb * (4 * NT);
  int mt   = rem2 / NT;
  int nt   = rem2 - mt * NT;

  const float* W  = (mat == 0) ? wq : (mat == 1) ? wk : wv;
  const float* Bs = (mat == 0) ? bq : (mat == 1) ? bk : bv;
  float*       Y  = (mat == 0) ? q  : (mat == 1) ? k  : v;

  int half = lane >> 4;
  int lm   = lane & 15;
  int row  = mt * 16 + lm;            // M index (output channel)
  int col  = nt * 16 + lm;            // N index (spatial)
  const float* xb = x + (size_t)bb * CCH * CHW;

  v8f acc = {};
#pragma unroll
  for (int kt = 0; kt < CCH / 4; ++kt) {
    int kb = kt * 4 + 2 * half;
    v2f a, b;
    a.x = W[row * CCH + kb];
    a.y = W[row * CCH + kb + 1];
    b.x = xb[(size_t)kb * CHW + col];
    b.y = xb[(size_t)(kb + 1) * CHW + col];
    acc = __builtin_amdgcn_wmma_f32_16x16x4_f32(false, a, false, b,
                                                (short)0, acc, false, false);
  }
  float* Yb = Y + (size_t)bb * CCH * CHW;
#pragma unroll
  for (int r = 0; r < 8; ++r) {
    int m = mt * 16 + r + 8 * half;
    Yb[(size_t)m * CHW + col] = acc[r] + Bs[m];
  }
}

// ---------------------------------------------------------------------------
// Fused attention. Reproduces the reference's *contiguous reinterpretation*:
//   uk[bi,di,ki,ci,s] = k_att[n=di%8, hm=(ki*16+ci)/49, depth=2*bi+di/8,
//                             tap=(ki*16+ci)%49 reflected at s]
// urpe collapses to P(hm,ry,rx) (d-independent), so
//   logit[ki] = qpe + sum_d q'_d * K_gather[d] - P * sum_d q'_d
// One thread per (bi, ci, s); softmax over the 49 taps; then the same scrambled
// gather on V.  Writes rate1 * out_att into d_out (full overwrite).
// ---------------------------------------------------------------------------
__device__ __forceinline__ int refl(int t, int n) {
  t = (t < 0) ? -t : t;
  return (t >= n) ? (2 * n - 2 - t) : t;
}

__global__ void attn_fused(const float* __restrict__ q,
                           const float* __restrict__ k,
                           const float* __restrict__ v,
                           const float* __restrict__ wp,   // (16,2)
                           const float* __restrict__ bp,   // (16,)
                           const float* __restrict__ rate1p,
                           float* __restrict__ out)        // (2,64,16,400)
{
  int gid = blockIdx.x * blockDim.x + threadIdx.x;  // 8*16*400 = 51200
  int s  = gid % HWSP;
  int ci = (gid / HWSP) % CD;
  int bi = gid / (HWSP * CD);                       // 0..7
  int y = s / WW, x = s % WW;

  float wpx[HD], wpy[HD], bpv[HD];
#pragma unroll
  for (int d = 0; d < HD; ++d) { wpx[d] = wp[2*d]; wpy[d] = wp[2*d+1]; bpv[d] = bp[d]; }

  const float lw = -1.f + 2.f * (float)x / (float)(WW - 1);
  const float lh = -1.f + 2.f * (float)y / (float)(HH - 1);

  const int qbase = (bi >> 2) * CCH + (bi & 3) * HD;
  float qv[HD];
  float qpe = 0.f, qsum = 0.f;
#pragma unroll
  for (int d = 0; d < HD; ++d) {
    qv[d] = q[(size_t)(qbase + d) * CHW + ci * HWSP + s] * 0.25f;  // scaling hd^-0.5
    qsum += qv[d];
    qpe  += qv[d] * (wpx[d] * lw + wpy[d] * lh + bpv[d]);
  }

  // scrambled per-d gather base offsets (shared by K and V passes)
  int off[HD];
#pragma unroll
  for (int d = 0; d < HD; ++d) {
    int n = d & 7;                       // batch*head source index
    int z = 2 * bi + (d >> 3);           // depth slice
    off[d] = ((n >> 2) * CCH + (n & 3) * HD) * CHW + z * HWSP;
  }

  float logit[KKA];
  float mx = -3.0e38f;
  for (int ki = 0; ki < KKA; ++ki) {
    int m2  = ki * HD + ci;
    int hm  = m2 / KKA;
    int tap = m2 - hm * KKA;
    int ry = refl(y + tap / KATAP - PADR, HH);
    int rx = refl(x + tap % KATAP - PADR, WW);
    float lw2 = -1.f + 2.f * (float)rx / (float)(WW - 1);
    float lh2 = -1.f + 2.f * (float)ry / (float)(HH - 1);
    float P = wpx[hm] * lw2 + wpy[hm] * lh2 + bpv[hm];
    int hoff = hm * CHW + ry * WW + rx;
    float acc = qpe - P * qsum;
#pragma unroll
    for (int d = 0; d < HD; ++d) acc += qv[d] * k[off[d] + hoff];
    logit[ki] = acc;
    mx = fmaxf(mx, acc);
  }
  float sum = 0.f;
  for (int ki = 0; ki < KKA; ++ki) { float e = expf(logit[ki] - mx); logit[ki] = e; sum += e; }
  float inv = 1.f / sum;

  float o[HD];
#pragma unroll
  for (int d = 0; d < HD; ++d) o[d] = 0.f;
  for (int ki = 0; ki < KKA; ++ki) {
    int m2  = ki * HD + ci;
    int hm  = m2 / KKA;
    int tap = m2 - hm * KKA;
    int ry = refl(y + tap / KATAP - PADR, HH);
    int rx = refl(x + tap % KATAP - PADR, WW);
    int hoff = hm * CHW + ry * WW + rx;
    float wgt = logit[ki] * inv;
#pragma unroll
    for (int d = 0; d < HD; ++d) o[d] += wgt * v[off[d] + hoff];
  }
  float r1 = rate1p[0];
#pragma unroll
  for (int d = 0; d < HD; ++d)
    out[(size_t)(qbase + d) * CHW + ci * HWSP + s] = r1 * o[d];
}

// ---------------------------------------------------------------------------
// fc mixing: f_conv[b, d*9+o, c, s] = sum_t w_fc[o,t] * qkv[b, t, d, c, s]
// qkv[t] -> q (t<4, head=t), k (4<=t<8), v (t>=8); channel = head*16 + d.
// ---------------------------------------------------------------------------
__global__ void fc_proj(const float* __restrict__ q, const float* __restrict__ k,
                        const float* __restrict__ v, const float* __restrict__ wfc,
                        float* __restrict__ fconv)
{
  int gid = blockIdx.x * blockDim.x + threadIdx.x;   // 2*144*6400 = 1,843,200
  int s  = gid % CHW;
  int ch = (gid / CHW) % (HD * 9);
  int bb = gid / (CHW * HD * 9);
  int d = ch / 9;
  int o = ch - 9 * d;
  float acc = 0.f;
#pragma unroll
  for (int t = 0; t < 12; ++t) {
    const float* src = (t < 4) ? q : (t < 8) ? k : v;
    acc += wfc[o * 12 + t] * src[(size_t)(bb * CCH + (t & 3) * HD + d) * CHW + s];
  }
  fconv[gid] = acc;
}

// ---------------------------------------------------------------------------
// Grouped 3x3x3 conv (groups=16, 9 in / 4 out per group, zero padding 1).
// Adds rate2 * out_conv into d_out.
// ---------------------------------------------------------------------------
__global__ void dep_conv(const float* __restrict__ fconv,  // (2,144,16,20,20)
                         const float* __restrict__ wdep,   // (64,9,3,3,3)
                         const float* __restrict__ bdep,
                         const float* __restrict__ rate2p,
                         float* __restrict__ out)
{
  int gid = blockIdx.x * blockDim.x + threadIdx.x;   // 2*64*6400 = 819,200
  int sp = gid % HWSP;
  int cc = (gid / HWSP) % CD;
  int oc = (gid / (HWSP * CD)) % CCH;
  int bb = gid / (HWSP * CD * CCH);
  int g  = oc >> 2;                                  // 4 output channels per group
  int y = sp / WW, x = sp % WW;

  float acc = bdep[oc];
  for (int o = 0; o < 9; ++o) {
    const float* fc = fconv + (size_t)(bb * (HD * 9) + g * 9 + o) * CHW;
    const float* wd = wdep + (size_t)(oc * 9 + o) * 27;
#pragma unroll
    for (int kd = 0; kd < 3; ++kd) {
      int zc = cc + kd - 1;
      if ((unsigned)zc >= (unsigned)CD) continue;
#pragma unroll
      for (int kh = 0; kh < 3; ++kh) {
        int zy = y + kh - 1;
        if ((unsigned)zy >= (unsigned)HH) continue;
#pragma unroll
        for (int kw = 0; kw < 3; ++kw) {
          int zx = x + kw - 1;
          if ((unsigned)zx >= (unsigned)WW) continue;
          acc += wd[(kd * 3 + kh) * 3 + kw] * fc[zc * HWSP + zy * WW + zx];
        }
      }
    }
  }
  out[gid] += rate2p[0] * acc;
}

// ---------------------------------------------------------------------------
extern "C" void kernel_launch(void* const* d_in, const int* in_sizes, int n_in,
                              void* d_out, int out_size, void* d_ws, size_t ws_size,
                              hipStream_t stream)
{
  (void)in_sizes; (void)n_in; (void)out_size; (void)ws_size;
  const float* x     = (const float*)d_in[0];
  const float* w_q   = (const float*)d_in[1];
  const float* b_q   = (const float*)d_in[2];
  const float* w_k   = (const float*)d_in[3];
  const float* b_k   = (const float*)d_in[4];
  const float* w_v   = (const float*)d_in[5];
  const float* b_v   = (const float*)d_in[6];
  const float* w_p   = (const float*)d_in[7];
  const float* b_p   = (const float*)d_in[8];
  const float* w_fc  = (const float*)d_in[9];
  const float* w_dep = (const float*)d_in[10];
  const float* b_dep = (const float*)d_in[11];
  const float* rate1 = (const float*)d_in[12];
  const float* rate2 = (const float*)d_in[13];
  float* out = (float*)d_out;
  float* ws  = (float*)d_ws;

  const size_t QKV = (size_t)NB * CCH * CHW;         // 819,200 floats each
  float* q  = ws;
  float* k  = ws + QKV;
  float* v  = ws + 2 * QKV;
  float* fc = ws + 3 * QKV;                          // 1,843,200 floats

  // 3 mats * 2 b * 4 Mtiles * 400 Ntiles = 9600 waves = 307,200 threads
  qkv_proj_wmma<<<1200, 256, 0, stream>>>(x, w_q, b_q, w_k, b_k, w_v, b_v, q, k, v);
  attn_fused  <<< 200, 256, 0, stream>>>(q, k, v, w_p, b_p, rate1, out);
  fc_proj     <<<7200, 256, 0, stream>>>(q, k, v, w_fc, fc);
  dep_conv    <<<3200, 256, 0, stream>>>(fc, w_dep, b_dep, rate2, out);
}